// ChamferDist_3006477107867
// MI455X (gfx1250) — compile-verified
//
#include <hip/hip_runtime.h>
#include <hip/hip_bf16.h>

typedef __attribute__((ext_vector_type(2))) float v2f;
typedef __attribute__((ext_vector_type(8))) float v8f;

#define CH_B 4
#define CH_N 8192
#define CH_M 8192
#define WG_THREADS 256
#define WG_WAVES 8
#define ROWS_PER_WG (WG_WAVES * 16)          // 128 rows per workgroup
#define WGS_PER_BATCH (CH_N / ROWS_PER_WG)   // 64
#define COL_CHUNK 4096                       // columns staged in LDS at a time (64 KB)

__global__ void __launch_bounds__(256) chamfer_init(float* __restrict__ colmin,
                                                    float* __restrict__ rowsum) {
    int i = blockIdx.x * 256 + threadIdx.x;
    if (i < CH_B * CH_M) colmin[i] = __builtin_inff();
    if (i < CH_B) rowsum[i] = 0.0f;
}

__global__ void __launch_bounds__(WG_THREADS)
chamfer_main(const float* __restrict__ adv, const float* __restrict__ ori,
             float* __restrict__ colmin, float* __restrict__ rowsum) {
    __shared__ float4 s_y[COL_CHUNK];        // 64 KB: {-2y0,-2y1,-2y2, |y|^2} per column
    __shared__ float  s_colmin[CH_M];        // 32 KB: per-WG column mins
    __shared__ float  s_x2[ROWS_PER_WG];     // x^2 for this WG's rows

    const int wg   = blockIdx.x;             // 0 .. B*WGS_PER_BATCH-1
    const int b    = wg / WGS_PER_BATCH;
    const int rblk = wg % WGS_PER_BATCH;
    const int tid  = threadIdx.x;
    const int wave = tid >> 5;
    const int lane = tid & 31;
    const int lo16 = lane & 15;
    const bool hi  = lane >= 16;

    // init per-WG column mins
    for (int i = tid; i < CH_M; i += WG_THREADS) s_colmin[i] = __builtin_inff();

    const float* __restrict__ xb = adv + (size_t)b * CH_N * 3;
    const float* __restrict__ yb = ori + (size_t)b * CH_M * 3;

    // This wave's 16-row strip of the adv (x) points
    const int row0  = rblk * ROWS_PER_WG + wave * 16;
    const int myrow = row0 + lo16;
    const float xx = xb[myrow * 3 + 0];
    const float xy = xb[myrow * 3 + 1];
    const float xz = xb[myrow * 3 + 2];
    const float x2 = xx * xx + xy * xy + xz * xz;
    if (!hi) s_x2[wave * 16 + lo16] = x2;
    __syncthreads();

    // A matrix (16x4 f32, K=3 padded): lanes 0-15 hold K=0,1; lanes 16-31 hold K=2,3
    v2f a;
    a.x = hi ? xz : xx;
    a.y = hi ? 0.0f : xy;

    // x^2 for the 8 C/D rows this lane touches (row = r + 8*hi)
    float x2v[8];
#pragma unroll
    for (int r = 0; r < 8; ++r) x2v[r] = s_x2[wave * 16 + (hi ? 8 : 0) + r];

    // Running row-min accumulator (elementwise over all column tiles)
    v8f acc;
#pragma unroll
    for (int r = 0; r < 8; ++r) acc[r] = __builtin_inff();

    for (int chunk = 0; chunk < CH_M; chunk += COL_CHUNK) {
        __syncthreads();   // previous chunk fully consumed before restage
        // Stage columns [chunk, chunk+COL_CHUNK): precompute B' operand and |y|^2
        for (int c = tid; c < COL_CHUNK; c += WG_THREADS) {
            const int col = chunk + c;
            const float y0 = yb[col * 3 + 0];
            const float y1 = yb[col * 3 + 1];
            const float y2 = yb[col * 3 + 2];
            float4 v;
            v.x = -2.0f * y0;
            v.y = -2.0f * y1;
            v.z = -2.0f * y2;
            v.w = y0 * y0 + y1 * y1 + y2 * y2;
            s_y[c] = v;
        }
        __syncthreads();

        for (int t = 0; t < COL_CHUNK; t += 16) {
            const float4 yv = s_y[t + lo16];           // ds_load_b128

            // B' = -2*y (4x16 f32, K padded): lanes 0-15 K=0,1; lanes 16-31 K=2,3
            v2f bv;
            bv.x = hi ? yv.z : yv.x;
            bv.y = hi ? 0.0f : yv.y;

            // C[i][j] = x2_i + y2_j  ->  D = full squared-distance tile
            v8f c;
#pragma unroll
            for (int r = 0; r < 8; ++r) c[r] = x2v[r] + yv.w;

            v8f d = __builtin_amdgcn_wmma_f32_16x16x4_f32(
                /*neg_a=*/false, a, /*neg_b=*/false, bv,
                /*c_mod=*/(short)0, c, /*reuse_a=*/false, /*reuse_b=*/false);

            // Row running min (per lane, elementwise over column tiles)
#pragma unroll
            for (int r = 0; r < 8; ++r) acc[r] = fminf(acc[r], d[r]);

            // Column min: reduce this lane's 8 rows, all 32 lanes atomically
            // min into LDS (both halves hit the same 16 addresses).
            float cm = fminf(d[0], d[1]);
#pragma unroll
            for (int r = 2; r < 8; ++r) cm = fminf(cm, d[r]);
            __hip_atomic_fetch_min(&s_colmin[chunk + t + lo16], cm,
                                   __ATOMIC_RELAXED, __HIP_MEMORY_SCOPE_WORKGROUP);
        }
    }

    // Row mins: reduce each acc element across the 16 lanes of each half-wave
#pragma unroll
    for (int r = 0; r < 8; ++r) {
        float v = acc[r];
        v = fminf(v, __shfl_xor(v, 1, 32));
        v = fminf(v, __shfl_xor(v, 2, 32));
        v = fminf(v, __shfl_xor(v, 4, 32));
        v = fminf(v, __shfl_xor(v, 8, 32));
        acc[r] = v;
    }
    float s = 0.0f;
#pragma unroll
    for (int r = 0; r < 8; ++r) s += acc[r];   // lo half: rows r; hi half: rows r+8
    s += __shfl_xor(s, 16, 32);                // total over the wave's 16 rows
    if (lane == 0) atomicAdd(&rowsum[b], s);

    __syncthreads();
    // Combine per-WG column mins into the global per-batch column mins
    for (int i = tid; i < CH_M; i += WG_THREADS)
        __hip_atomic_fetch_min(&colmin[(size_t)b * CH_M + i], s_colmin[i],
                               __ATOMIC_RELAXED, __HIP_MEMORY_SCOPE_AGENT);
}

__global__ void __launch_bounds__(256)
chamfer_final(const float* __restrict__ colmin, const float* __restrict__ rowsum,
              const float* __restrict__ weights, float* __restrict__ out) {
    __shared__ float red[256];
    __shared__ float total_s;
    if (threadIdx.x == 0) total_s = 0.0f;
    __syncthreads();
    for (int b = 0; b < CH_B; ++b) {
        float ls = 0.0f;
        for (int i = threadIdx.x; i < CH_M; i += 256)
            ls += colmin[(size_t)b * CH_M + i];
        red[threadIdx.x] = ls;
        __syncthreads();
        for (int off = 128; off > 0; off >>= 1) {
            if (threadIdx.x < off) red[threadIdx.x] += red[threadIdx.x + off];
            __syncthreads();
        }
        if (threadIdx.x == 0) {
            total_s += weights[b] * (rowsum[b] * (1.0f / (float)CH_N) +
                                     red[0]    * (1.0f / (float)CH_M));
        }
        __syncthreads();
    }
    if (threadIdx.x == 0) out[0] = total_s * (1.0f / (float)CH_B);
}

extern "C" void kernel_launch(void* const* d_in, const int* in_sizes, int n_in,
                              void* d_out, int out_size, void* d_ws, size_t ws_size,
                              hipStream_t stream) {
    const float* adv     = (const float*)d_in[0];  // [B,N,3]
    const float* ori     = (const float*)d_in[1];  // [B,M,3]
    const float* weights = (const float*)d_in[2];  // [B]
    float* out = (float*)d_out;

    float* colmin = (float*)d_ws;                                 // B*M floats (128 KB)
    float* rowsum = (float*)((char*)d_ws + (size_t)CH_B * CH_M * sizeof(float));

    chamfer_init<<<(CH_B * CH_M + 255) / 256, 256, 0, stream>>>(colmin, rowsum);
    chamfer_main<<<CH_B * WGS_PER_BATCH, WG_THREADS, 0, stream>>>(adv, ori, colmin, rowsum);
    chamfer_final<<<1, 256, 0, stream>>>(colmin, rowsum, weights, out);
}